// MultiScaleTriplane_4406636446010
// MI455X (gfx1250) — compile-verified
//
#include <hip/hip_runtime.h>

typedef float v4f __attribute__((ext_vector_type(4)));

#define DEV __device__ __forceinline__

namespace {

constexpr int CH = 64;
constexpr int W0 = 256, W1 = 128, W2 = 64;
constexpr int HW0 = W0 * W0, HW1 = W1 * W1, HW2 = W2 * W2;
constexpr size_t L0P = (size_t)HW0 * CH;   // floats per level-0 plane (HWC)
constexpr size_t L1P = (size_t)HW1 * CH;
constexpr size_t L2P = (size_t)HW2 * CH;
constexpr size_t WS_FLOATS = 3 * (L0P + L1P + L2P);   // 16,515,072 floats = 63 MiB

// Keys cubic-convolution weights, a = -0.75, taps at offsets [-1,0,1,2], t in [0,1)
DEV void cubicw(float t, float w[4]) {
  const float A = -0.75f;
  float x = t + 1.0f;                                     // 1 <= x < 2
  w[0] = ((A * x - 5.0f * A) * x + 8.0f * A) * x - 4.0f * A;
  w[1] = ((A + 2.0f) * t - (A + 3.0f)) * t * t + 1.0f;    // |t| <= 1
  float u = 1.0f - t;
  w[2] = ((A + 2.0f) * u - (A + 3.0f)) * u * u + 1.0f;
  float v = 2.0f - t;
  w[3] = ((A * v - 5.0f * A) * v + 8.0f * A) * v - 4.0f * A;
}

// Bicubic sample of one (square WxW, 64-ch) plane, accumulating 4 channels/lane.
// TR=true : plane stored (H,W,C)  -> one 128-bit coalesced load per tap
// TR=false: plane stored (C,H,W)  -> 4 strided scalar loads per tap (fallback)
template <bool TR>
DEV void samp(const float* __restrict__ pl, int W, float gx, float gy, int ch,
              v4f& acc) {
  const float fx = (gx + 1.0f) * 0.5f * (float)(W - 1);
  const float fy = (gy + 1.0f) * 0.5f * (float)(W - 1);
  const float bx = floorf(fx), by = floorf(fy);
  float wx[4], wy[4];
  cubicw(fx - bx, wx);
  cubicw(fy - by, wy);
  const int xb = (int)bx - 1, yb = (int)by - 1;
  int ix[4], iy[4];
#pragma unroll
  for (int i = 0; i < 4; ++i) {
    ix[i] = min(max(xb + i, 0), W - 1);   // border clamp
    iy[i] = min(max(yb + i, 0), W - 1);
  }
  const int HW = W * W;
#pragma unroll
  for (int j = 0; j < 4; ++j) {
    const int rb = iy[j] * W;
    const float wyj = wy[j];
    if constexpr (TR) {
      // all 32-bit offsets; issue the 4 row taps together, then FMA
      const float* rowp = pl + (rb * CH + ch);
      const v4f v0 = *reinterpret_cast<const v4f*>(rowp + ix[0] * CH);
      const v4f v1 = *reinterpret_cast<const v4f*>(rowp + ix[1] * CH);
      const v4f v2 = *reinterpret_cast<const v4f*>(rowp + ix[2] * CH);
      const v4f v3 = *reinterpret_cast<const v4f*>(rowp + ix[3] * CH);
      acc += (wyj * wx[0]) * v0;
      acc += (wyj * wx[1]) * v1;
      acc += (wyj * wx[2]) * v2;
      acc += (wyj * wx[3]) * v3;
    } else {
#pragma unroll
      for (int i = 0; i < 4; ++i) {
        const float w = wyj * wx[i];
        const float* q = pl + (ch * HW + rb + ix[i]);
        v4f v = {q[0], q[HW], q[2 * HW], q[3 * HW]};
        acc += w * v;
      }
    }
  }
}

DEV void nt_store4(float* p, v4f v) {
  // streaming output: keep the plane data resident in L2 instead
  __builtin_nontemporal_store(v, reinterpret_cast<v4f*>(p));
}

DEV v4f nt_load4(const float* p) {
  return __builtin_nontemporal_load(
      const_cast<v4f*>(reinterpret_cast<const v4f*>(p)));
}

}  // namespace

// (C,H,W) -> (H,W,C) via LDS tile; input read non-temporally (read-once).
// Fully unrolled: 4 independent b128 NT loads in flight per thread.
__global__ __launch_bounds__(256) void transpose_chw_hwc(
    const float* __restrict__ in, float* __restrict__ out, int HW) {
  __shared__ float tile[CH][65];   // +1 pad: conflict-free both phases
  const int s0 = (int)blockIdx.x << 6;  // 64 spatial positions per block
  const int tid = (int)threadIdx.x;

  v4f r[4];
#pragma unroll
  for (int k = 0; k < 4; ++k) {
    const int i = tid + k * 256;          // 0..1023
    const int c = i >> 4, sg = (i & 15) << 2;
    r[k] = nt_load4(in + (c * HW + s0 + sg));   // 32-bit index math
  }
#pragma unroll
  for (int k = 0; k < 4; ++k) {
    const int i = tid + k * 256;
    const int c = i >> 4, sg = (i & 15) << 2;
    tile[c][sg + 0] = r[k].x; tile[c][sg + 1] = r[k].y;
    tile[c][sg + 2] = r[k].z; tile[c][sg + 3] = r[k].w;
  }
  __syncthreads();
#pragma unroll
  for (int k = 0; k < 4; ++k) {
    const int i = tid + k * 256;
    const int s = i >> 4, cg = (i & 15) << 2;
    v4f v = {tile[cg + 0][s], tile[cg + 1][s], tile[cg + 2][s], tile[cg + 3][s]};
    *reinterpret_cast<v4f*>(out + ((s0 + s) * CH + cg)) = v;  // RT: stay in L2
  }
}

// One wave32 handles 2 points: 16 lanes/point, 4 channels/lane (b128 loads).
template <bool TR>
__global__ __launch_bounds__(256) void tri_sample(
    const float* __restrict__ coords,
    const float* __restrict__ px0, const float* __restrict__ py0, const float* __restrict__ pz0,
    const float* __restrict__ px1, const float* __restrict__ py1, const float* __restrict__ pz1,
    const float* __restrict__ px2, const float* __restrict__ py2, const float* __restrict__ pz2,
    float* __restrict__ out, int N) {
  const int p = (int)blockIdx.x * 16 + ((int)threadIdx.x >> 4);
  if (p >= N) return;
  const int ch = ((int)threadIdx.x & 15) << 2;
  const float g0 = (coords[3 * p + 0] + 1.0f) * 0.5f;
  const float g1 = (coords[3 * p + 1] + 1.0f) * 0.5f;
  const float g2 = (coords[3 * p + 2] + 1.0f) * 0.5f;
  float* o = out + (p * (3 * CH) + ch);
  {
    v4f acc = {0.f, 0.f, 0.f, 0.f};
    samp<TR>(px0, W0, g0, g1, ch, acc);
    samp<TR>(py0, W0, g1, g2, ch, acc);
    samp<TR>(pz0, W0, g0, g2, ch, acc);
    nt_store4(o, acc);
  }
  {
    v4f acc = {0.f, 0.f, 0.f, 0.f};
    samp<TR>(px1, W1, g0, g1, ch, acc);
    samp<TR>(py1, W1, g1, g2, ch, acc);
    samp<TR>(pz1, W1, g0, g2, ch, acc);
    nt_store4(o + CH, acc);
  }
  {
    v4f acc = {0.f, 0.f, 0.f, 0.f};
    samp<TR>(px2, W2, g0, g1, ch, acc);
    samp<TR>(py2, W2, g1, g2, ch, acc);
    samp<TR>(pz2, W2, g0, g2, ch, acc);
    nt_store4(o + 2 * CH, acc);
  }
}

extern "C" void kernel_launch(void* const* d_in, const int* in_sizes, int n_in,
                              void* d_out, int out_size, void* d_ws, size_t ws_size,
                              hipStream_t stream) {
  const float* coords = (const float*)d_in[0];
  const float* pin[9];
  for (int i = 0; i < 9; ++i) pin[i] = (const float*)d_in[1 + i];
  float* out = (float*)d_out;
  const int N = in_sizes[0] / 3;

  static const int hw[9] = {HW0, HW0, HW0, HW1, HW1, HW1, HW2, HW2, HW2};

  if (ws_size >= WS_FLOATS * sizeof(float)) {
    float* ws = (float*)d_ws;
    size_t off[9];
    size_t o = 0;
    for (int i = 0; i < 9; ++i) { off[i] = o; o += (size_t)hw[i] * CH; }
    for (int i = 0; i < 9; ++i)
      transpose_chw_hwc<<<hw[i] / 64, 256, 0, stream>>>(pin[i], ws + off[i], hw[i]);
    tri_sample<true><<<(N + 15) / 16, 256, 0, stream>>>(
        coords, ws + off[0], ws + off[1], ws + off[2], ws + off[3], ws + off[4],
        ws + off[5], ws + off[6], ws + off[7], ws + off[8], out, N);
  } else {
    // workspace too small: correct (slower) CHW-layout path
    tri_sample<false><<<(N + 15) / 16, 256, 0, stream>>>(
        coords, pin[0], pin[1], pin[2], pin[3], pin[4], pin[5], pin[6], pin[7],
        pin[8], out, N);
  }
}